// GNNLayer_76244259439107
// MI455X (gfx1250) — compile-verified
//
#include <hip/hip_runtime.h>
#include <hip/hip_bf16.h>
#include <math.h>

typedef __attribute__((ext_vector_type(2))) float v2f;
typedef __attribute__((ext_vector_type(8))) float v8f;

#define D_DIM 64

// ---------------------------------------------------------------------------
// Zero-fill (agg must be zeroed every call; harness does not re-poison ws)
// ---------------------------------------------------------------------------
__global__ void gnn_zero_kernel(float* __restrict__ p, long n) {
  long i = (long)blockIdx.x * blockDim.x + threadIdx.x;
  if (i < n) p[i] = 0.0f;
}

// ---------------------------------------------------------------------------
// C[M x 64] = gatherRows(A, rowIdx)[M x 64] @ W[64 x 64] (+ bias)
// One wave computes a 16-row x 64-col slab using V_WMMA_F32_16X16X4_F32.
// fp32 A/B keeps exact reference precision; K advances 4 per WMMA.
//
// VGPR layouts (CDNA5 ISA 7.12.2):
//   A 16x4  : lane = M + 16*(K>=2), vgpr = K&1
//   B 4x16  : lane = N + 16*(K>=2), vgpr = K&1
//   C 16x16 : vgpr i -> row i (lanes 0-15) / row i+8 (lanes 16-31), col = lane&15
// ---------------------------------------------------------------------------
__global__ void gnn_wmma_matmul64(const float* __restrict__ A,
                                  const int*   __restrict__ rowIdx,
                                  const float* __restrict__ W,
                                  const float* __restrict__ bias,
                                  float* __restrict__ C, int M)
{
  int wave = (int)(((unsigned)blockIdx.x * blockDim.x + threadIdx.x) >> 5);
  int lane = threadIdx.x & 31;
  int m0 = wave * 16;
  if (m0 >= M) return;                 // wave-uniform exit: EXEC stays all-1s

  int mr = lane & 15;                  // row (A) / col (B) within tile
  int hi = lane >> 4;                  // K-half selector
  int m  = m0 + mr;
  int mc = (m < M) ? m : (M - 1);      // clamp A row (masked at store)
  int arow = rowIdx ? rowIdx[mc] : mc; // uniform branch (kernel arg)
  const float* Arow = A + (size_t)arow * D_DIM;

  v8f acc0 = {}, acc1 = {}, acc2 = {}, acc3 = {};
  #pragma unroll
  for (int k = 0; k < D_DIM; k += 4) {
    int koff = k + 2 * hi;
    v2f a; a.x = Arow[koff]; a.y = Arow[koff + 1];
    const float* Wr0 = W + (size_t)koff * D_DIM;
    const float* Wr1 = Wr0 + D_DIM;
    v2f b0; b0.x = Wr0[mr +  0]; b0.y = Wr1[mr +  0];
    v2f b1; b1.x = Wr0[mr + 16]; b1.y = Wr1[mr + 16];
    v2f b2; b2.x = Wr0[mr + 32]; b2.y = Wr1[mr + 32];
    v2f b3; b3.x = Wr0[mr + 48]; b3.y = Wr1[mr + 48];
    acc0 = __builtin_amdgcn_wmma_f32_16x16x4_f32(false, a, false, b0, (short)0, acc0, false, false);
    acc1 = __builtin_amdgcn_wmma_f32_16x16x4_f32(false, a, false, b1, (short)0, acc1, false, false);
    acc2 = __builtin_amdgcn_wmma_f32_16x16x4_f32(false, a, false, b2, (short)0, acc2, false, false);
    acc3 = __builtin_amdgcn_wmma_f32_16x16x4_f32(false, a, false, b3, (short)0, acc3, false, false);
  }

  v8f accs[4] = {acc0, acc1, acc2, acc3};
  #pragma unroll
  for (int t = 0; t < 4; ++t) {
    int n = t * 16 + mr;
    float bv = bias ? bias[n] : 0.0f;
    #pragma unroll
    for (int i = 0; i < 8; ++i) {
      int row = m0 + i + 8 * hi;
      if (row < M) C[(size_t)row * D_DIM + n] = accs[t][i] + bv;
    }
  }
}

// ---------------------------------------------------------------------------
// Per-edge pass: one wave per edge; lane handles components d and d+32.
//   pre  = relu(HsW[sub] + HrW[rel] + HqrW[r_idx])          (table adds)
//   alpha= sigmoid(dot(pre, w_alpha) + b_alpha)             (shuffle-reduce)
//   atomicAdd(agg[obj], alpha * (hidden[sub] + rela[rel]))
// All gather/scatter tables are L2-resident on MI455X (192 MB L2).
// ---------------------------------------------------------------------------
__global__ void gnn_edge_kernel(const int*   __restrict__ edges,
                                const float* __restrict__ hidden,
                                const float* __restrict__ rela,
                                const float* __restrict__ HsW,
                                const float* __restrict__ HrW,
                                const float* __restrict__ HqrW,
                                const float* __restrict__ w_alpha,
                                const float* __restrict__ b_alpha,
                                float* __restrict__ agg, int E)
{
  int wave = (int)(((unsigned)blockIdx.x * blockDim.x + threadIdx.x) >> 5);
  int lane = threadIdx.x & 31;
  if (wave >= E) return;

  const int* e = edges + (size_t)wave * 6;
  // prefetch a few edges ahead in the streaming dimension
  __builtin_prefetch((const void*)(e + 6 * 64), 0, 0);

  int ridx = e[0];
  int rel  = e[2];
  int sub  = e[4];
  int obj  = e[5];

  size_t sB = (size_t)sub  * D_DIM;
  size_t rB = (size_t)rel  * D_DIM;
  size_t qB = (size_t)ridx * D_DIM;

  float p0 = HsW[sB + lane]      + HrW[rB + lane]      + HqrW[qB + lane];
  float p1 = HsW[sB + lane + 32] + HrW[rB + lane + 32] + HqrW[qB + lane + 32];
  p0 = fmaxf(p0, 0.0f);
  p1 = fmaxf(p1, 0.0f);

  float part = p0 * w_alpha[lane] + p1 * w_alpha[lane + 32];
  #pragma unroll
  for (int off = 16; off > 0; off >>= 1)
    part += __shfl_xor(part, off, 32);

  float alpha = 1.0f / (1.0f + __expf(-(part + b_alpha[0])));

  float m0 = alpha * (hidden[sB + lane]      + rela[rB + lane]);
  float m1 = alpha * (hidden[sB + lane + 32] + rela[rB + lane + 32]);

  size_t oB = (size_t)obj * D_DIM;
  atomicAdd(&agg[oB + lane],      m0);
  atomicAdd(&agg[oB + lane + 32], m1);
}

// ---------------------------------------------------------------------------
// Launch
// Inputs (setup_inputs order):
//  0 q_rel(B int) 1 hidden(N*D f32) 2 edges(E*6 int) 3 rela_embed(R*D f32)
//  4 Ws_attn 5 Wr_attn 6 Wqr_attn (D*A f32) 7 b_qr(A) 8 w_alpha(A) 9 b_alpha(1)
// 10 W_h(D*D f32) 11 n_node(1 int)
// Output: (N, D) f32.
// ---------------------------------------------------------------------------
extern "C" void kernel_launch(void* const* d_in, const int* in_sizes, int n_in,
                              void* d_out, int out_size, void* d_ws, size_t ws_size,
                              hipStream_t stream) {
  const int*   q_rel   = (const int*)  d_in[0];
  const float* hidden  = (const float*)d_in[1];
  const int*   edges   = (const int*)  d_in[2];
  const float* rela    = (const float*)d_in[3];
  const float* Ws      = (const float*)d_in[4];
  const float* Wr      = (const float*)d_in[5];
  const float* Wqr     = (const float*)d_in[6];
  const float* b_qr    = (const float*)d_in[7];
  const float* w_alpha = (const float*)d_in[8];
  const float* b_alpha = (const float*)d_in[9];
  const float* W_h     = (const float*)d_in[10];
  float* out = (float*)d_out;

  const int N = in_sizes[1] / D_DIM;   // 50000
  const int E = in_sizes[2] / 6;       // 1.6M
  const int R = in_sizes[3] / D_DIM;   // 401
  const int B = in_sizes[0];           // 64

  // Workspace layout (floats): agg[N*64] | HsW[N*64] | HrW[R*64] | HqrW[B*64]
  float* agg  = (float*)d_ws;
  float* HsW  = agg  + (size_t)N * D_DIM;
  float* HrW  = HsW  + (size_t)N * D_DIM;
  float* HqrW = HrW  + (size_t)R * D_DIM;

  // 1) zero agg
  long aggN = (long)N * D_DIM;
  gnn_zero_kernel<<<(unsigned)((aggN + 255) / 256), 256, 0, stream>>>(agg, aggN);

  auto gridFor = [](int M) -> unsigned {
    long threads = (long)((M + 15) / 16) * 32;
    return (unsigned)((threads + 255) / 256);
  };

  // 2) precompute projection tables (fp32 WMMA)
  gnn_wmma_matmul64<<<gridFor(N), 256, 0, stream>>>(hidden, nullptr, Ws,  nullptr, HsW,  N);
  gnn_wmma_matmul64<<<gridFor(R), 256, 0, stream>>>(rela,   nullptr, Wr,  nullptr, HrW,  R);
  gnn_wmma_matmul64<<<gridFor(B), 256, 0, stream>>>(rela,   q_rel,   Wqr, b_qr,    HqrW, B);

  // 3) per-edge attention + scatter-add (1 wave per edge)
  long eThreads = (long)E * 32;
  gnn_edge_kernel<<<(unsigned)((eThreads + 255) / 256), 256, 0, stream>>>(
      edges, hidden, rela, HsW, HrW, HqrW, w_alpha, b_alpha, agg, E);

  // 4) out = agg @ W_h (fp32 WMMA)
  gnn_wmma_matmul64<<<gridFor(N), 256, 0, stream>>>(agg, nullptr, W_h, nullptr, out, N);
}